// MyTransformerXL_52974126629025
// MI455X (gfx1250) — compile-verified
//
#include <hip/hip_runtime.h>
#include <math.h>

#define LNUM   4
#define DMODEL 256
#define HEADS  8
#define DHEAD  32
#define DINNER 1024
#define SEGL   1024
#define MEML   1024
#define KLENL  2048
#define BATCH  2
#define NLEADS 12
#define LN_EPS 1e-5f

typedef float v2f __attribute__((ext_vector_type(2)));
typedef float v8f __attribute__((ext_vector_type(8)));

__device__ __forceinline__ v8f wmma4(v2f a, v2f b, v8f c) {
  // D(16x16,f32) = A(16x4,f32) * B(4x16,f32) + C
  return __builtin_amdgcn_wmma_f32_16x16x4_f32(false, a, false, b, (short)0, c,
                                               false, false);
}

// ---------------------------------------------------------------- pos emb
__global__ void k_pos_emb(float* __restrict__ r) {
  int row = blockIdx.x;
  int d = threadIdx.x;
  float pos = (float)(KLENL - 1 - row);
  int j = (d < DMODEL / 2) ? d : d - DMODEL / 2;
  float inv_freq = powf(10000.0f, -(2.0f * (float)j) / (float)DMODEL);
  float v = pos * inv_freq;
  r[(size_t)row * DMODEL + d] = (d < DMODEL / 2) ? sinf(v) : cosf(v);
}

// ------------------------------------------------------------ input proj
__global__ void k_input_proj(const float* __restrict__ src,
                             const float* __restrict__ W_in,
                             const float* __restrict__ b_in,
                             float* __restrict__ core,
                             float* __restrict__ mems0_out) {
  int row = blockIdx.x;          // i*B + b
  int d = threadIdx.x;
  int i = row / BATCH, b = row % BATCH;
  float acc = b_in[d];
#pragma unroll
  for (int l = 0; l < NLEADS; ++l)
    acc += src[((size_t)b * NLEADS + l) * SEGL + i] * W_in[l * DMODEL + d];
  core[(size_t)row * DMODEL + d] = acc;
  mems0_out[(size_t)row * DMODEL + d] = acc;   // new_mems[0] == hids[0]
}

// ------------------------------------------------------------------ copy
__global__ void k_copy(float* __restrict__ dst, const float* __restrict__ src,
                       int n) {
  int idx = blockIdx.x * blockDim.x + threadIdx.x;
  if (idx < n) dst[idx] = src[idx];
}

// ---------------------------------------------------- generic WMMA GEMM
// C[M x N] = A[M x K] @ B[K x N] (+bias, +relu). M%64==0, N%64==0, K%16==0.
__global__ void k_gemm(const float* __restrict__ A, const float* __restrict__ Bm,
                       const float* __restrict__ bias, float* __restrict__ C,
                       int M, int N, int K, int relu) {
  __shared__ float As[64][17];   // pad 17: conflict-free column reads
  __shared__ float Bs[16][64];
  const int n0 = blockIdx.x * 64, m0 = blockIdx.y * 64;
  const int t = threadIdx.x;
  const int lane = t & 31, wid = t >> 5;
  const int wm = (wid & 3) * 16;     // wave row sub-tile
  const int wn = (wid >> 2) * 32;    // wave col sub-tile (2 x 16 cols)
  const int half = lane >> 4, ln = lane & 15;
  const int koff = half * 2;
  v8f o0 = {0, 0, 0, 0, 0, 0, 0, 0};
  v8f o1 = {0, 0, 0, 0, 0, 0, 0, 0};

  for (int k0 = 0; k0 < K; k0 += 16) {
    __syncthreads();
    {
      int r = t >> 2, c = (t & 3) * 4;
      float4 va = *(const float4*)(A + (size_t)(m0 + r) * K + k0 + c);
      As[r][c + 0] = va.x; As[r][c + 1] = va.y;
      As[r][c + 2] = va.z; As[r][c + 3] = va.w;
      int rb = t >> 4, cb = (t & 15) * 4;
      float4 vb = *(const float4*)(Bm + (size_t)(k0 + rb) * N + n0 + cb);
      *(float4*)(&Bs[rb][cb]) = vb;
    }
    __syncthreads();
#pragma unroll
    for (int kk = 0; kk < 16; kk += 4) {
      v2f a, b0, b1;
      a.x = As[wm + ln][kk + koff];
      a.y = As[wm + ln][kk + koff + 1];
      b0.x = Bs[kk + koff][wn + ln];
      b0.y = Bs[kk + koff + 1][wn + ln];
      b1.x = Bs[kk + koff][wn + 16 + ln];
      b1.y = Bs[kk + koff + 1][wn + 16 + ln];
      o0 = wmma4(a, b0, o0);
      o1 = wmma4(a, b1, o1);
    }
  }
#pragma unroll
  for (int v = 0; v < 8; ++v) {
    int row = m0 + wm + v + 8 * half;
    int c0 = n0 + wn + ln, c1 = c0 + 16;
    float x0 = o0[v] + (bias ? bias[c0] : 0.0f);
    float x1 = o1[v] + (bias ? bias[c1] : 0.0f);
    if (relu) { x0 = fmaxf(x0, 0.0f); x1 = fmaxf(x1, 0.0f); }
    C[(size_t)row * N + c0] = x0;
    C[(size_t)row * N + c1] = x1;
  }
}

// -------------------------------------------- fused rel-attention (flash)
// One wave per (b, h, 16-row q tile). AC & P@V via WMMA; the rel_shift BD
// band (31 rk rows per j-tile) via LDS-staged VALU dots; online softmax
// with wave32 shfl-xor reductions over the 16-lane halves.
__global__ void __launch_bounds__(32)
k_attn(const float* __restrict__ heads, const float* __restrict__ rk,
       const float* __restrict__ rwb, const float* __restrict__ rrb,
       const unsigned char* __restrict__ mask, float* __restrict__ vec) {
  __shared__ float Qw[16][32];
  __shared__ float Qr[16][32];
  __shared__ float Ks[16][32];
  __shared__ float Vs[16][32];
  __shared__ float RKs[31][32];
  __shared__ float Ps[16][16];

  const int bid = blockIdx.x;
  const int it = bid & (SEGL / 16 - 1);
  const int bh = bid >> 6;
  const int b = bh & (BATCH - 1);
  const int h = bh >> 1;
  const int i0 = it * 16;

  const int lane = threadIdx.x;
  const int half = lane >> 4, ln = lane & 15;
  const int koff = half * 2;

  {  // Q tile + biases
    int r = lane >> 1;
    int c0 = (lane & 1) * 16;
    for (int c = 0; c < 16; ++c) {
      int cc = c0 + c;
      float q = heads[((size_t)(MEML + i0 + r) * BATCH + b) * 768 + h * DHEAD + cc];
      Qw[r][cc] = q + rwb[h * DHEAD + cc];
      Qr[r][cc] = q + rrb[h * DHEAD + cc];
    }
  }
  __syncthreads();

  float mprev[8], lsum[8];
  v8f o0 = {0, 0, 0, 0, 0, 0, 0, 0};
  v8f o1 = {0, 0, 0, 0, 0, 0, 0, 0};
#pragma unroll
  for (int v = 0; v < 8; ++v) { mprev[v] = -INFINITY; lsum[v] = 0.0f; }

  const float sc = 0.17677669529663687f;  // 1/sqrt(DHEAD)

  for (int j0 = 0; j0 < KLENL; j0 += 16) {
    __syncthreads();
    {  // stage K, V, and the rel-shift rk band
      int r = lane >> 1;
      int c0 = (lane & 1) * 16;
      const float* kp = heads + ((size_t)(j0 + r) * BATCH + b) * 768 + 256 + h * DHEAD;
      const float* vp = heads + ((size_t)(j0 + r) * BATCH + b) * 768 + 512 + h * DHEAD;
      for (int c = 0; c < 16; ++c) {
        Ks[r][c0 + c] = kp[c0 + c];
        Vs[r][c0 + c] = vp[c0 + c];
      }
      int bandlo = j0 + (SEGL - 1) - i0 - 15;  // m = bandlo + 15 + N - M
      for (int tt = r; tt < 31; tt += 16) {
        int m = bandlo + tt;
        bool ok = (m >= 0) && (m < KLENL);     // OOB rows == rel_shift zero pad
        for (int c = 0; c < 16; ++c)
          RKs[tt][c0 + c] = ok ? rk[(size_t)m * 256 + h * DHEAD + c0 + c] : 0.0f;
      }
    }
    __syncthreads();

    // S = Qw @ K^T  (K-dim = 32, eight 16x16x4 WMMAs)
    v8f s = {0, 0, 0, 0, 0, 0, 0, 0};
#pragma unroll
    for (int kk = 0; kk < 32; kk += 4) {
      v2f a, bb;
      a.x = Qw[ln][kk + koff];
      a.y = Qw[ln][kk + koff + 1];
      bb.x = Ks[ln][kk + koff];      // B[k][n] = K[n][k]
      bb.y = Ks[ln][kk + koff + 1];
      s = wmma4(a, bb, s);
    }

    // + BD band, scale, mask
#pragma unroll
    for (int v = 0; v < 8; ++v) {
      int M = v + 8 * half;
      int N = ln;
      float bd = 0.0f;
#pragma unroll 8
      for (int d = 0; d < 32; ++d) bd += Qr[M][d] * RKs[15 + N - M][d];
      float sv = (s[v] + bd) * sc;
      sv = mask[(size_t)(i0 + M) * KLENL + j0 + N] ? -INFINITY : sv;
      s[v] = sv;
    }

    // online softmax (rows live across 16-lane halves in C layout)
#pragma unroll
    for (int v = 0; v < 8; ++v) {
      float tm = s[v];
      tm = fmaxf(tm, __shfl_xor(tm, 1, 32));
      tm = fmaxf(tm, __shfl_xor(tm, 2, 32));
      tm = fmaxf(tm, __shfl_xor(tm, 4, 32));
      tm = fmaxf(tm, __shfl_xor(tm, 8, 32));
      float mnew = fmaxf(mprev[v], tm);
      float scale = (mprev[v] == -INFINITY) ? 0.0f : expf(mprev[v] - mnew);
      float p = expf(s[v] - mnew);
      float ts = p;
      ts += __shfl_xor(ts, 1, 32);
      ts += __shfl_xor(ts, 2, 32);
      ts += __shfl_xor(ts, 4, 32);
      ts += __shfl_xor(ts, 8, 32);
      lsum[v] = lsum[v] * scale + ts;
      o0[v] *= scale;
      o1[v] *= scale;
      mprev[v] = mnew;
      Ps[v + 8 * half][ln] = p;    // C-layout -> A-layout via LDS
    }
    __syncthreads();

    // O += P(16x16) @ V(16x32)
#pragma unroll
    for (int kk = 0; kk < 16; kk += 4) {
      v2f a, b0v, b1v;
      a.x = Ps[ln][kk + koff];
      a.y = Ps[ln][kk + koff + 1];
      b0v.x = Vs[kk + koff][ln];
      b0v.y = Vs[kk + koff + 1][ln];
      b1v.x = Vs[kk + koff][16 + ln];
      b1v.y = Vs[kk + koff + 1][16 + ln];
      o0 = wmma4(a, b0v, o0);
      o1 = wmma4(a, b1v, o1);
    }
  }

#pragma unroll
  for (int v = 0; v < 8; ++v) {
    float inv = 1.0f / lsum[v];
    int i = i0 + v + 8 * half;
    vec[((size_t)i * BATCH + b) * 256 + h * DHEAD + ln] = o0[v] * inv;
    vec[((size_t)i * BATCH + b) * 256 + h * DHEAD + 16 + ln] = o1[v] * inv;
  }
}

// ------------------------------------------------------- residual + LN
__global__ void k_add_ln(const float* __restrict__ x, const float* __restrict__ dl,
                         const float* __restrict__ g, const float* __restrict__ bb,
                         float* __restrict__ out_core, float* __restrict__ out_mems) {
  __shared__ float red[DMODEL];
  int row = blockIdx.x, t = threadIdx.x;
  float v = x[(size_t)row * DMODEL + t] + dl[(size_t)row * DMODEL + t];
  red[t] = v;
  __syncthreads();
  for (int s = DMODEL / 2; s > 0; s >>= 1) {
    if (t < s) red[t] += red[t + s];
    __syncthreads();
  }
  float mu = red[0] * (1.0f / DMODEL);
  __syncthreads();
  float dv = v - mu;
  red[t] = dv * dv;
  __syncthreads();
  for (int s = DMODEL / 2; s > 0; s >>= 1) {
    if (t < s) red[t] += red[t + s];
    __syncthreads();
  }
  float var = red[0] * (1.0f / DMODEL);
  float y = dv * rsqrtf(var + LN_EPS) * g[t] + bb[t];
  out_core[(size_t)row * DMODEL + t] = y;
  if (out_mems) out_mems[(size_t)row * DMODEL + t] = y;
}

// --------------------------------------------------------------- decoder
__global__ void k_decoder(const float* __restrict__ core,
                          const float* __restrict__ Wd,
                          const float* __restrict__ bd,
                          float* __restrict__ pred) {
  int idx = blockIdx.x * blockDim.x + threadIdx.x;
  if (idx >= BATCH * NLEADS * SEGL) return;
  int i = idx % SEGL;
  int l = (idx / SEGL) % NLEADS;
  int b = idx / (SEGL * NLEADS);
  float acc = bd[l];
  const float* cr = core + ((size_t)i * BATCH + b) * DMODEL;
  for (int d = 0; d < DMODEL; ++d) acc += cr[d] * Wd[d * NLEADS + l];
  pred[idx] = acc;
}

// ---------------------------------------------------------------- launch
extern "C" void kernel_launch(void* const* d_in, const int* in_sizes, int n_in,
                              void* d_out, int out_size, void* d_ws, size_t ws_size,
                              hipStream_t stream) {
  (void)in_sizes; (void)n_in; (void)out_size; (void)ws_size;
  const float* src = (const float*)d_in[0];
  const float* mems = (const float*)d_in[1];
  const unsigned char* attn_mask = (const unsigned char*)d_in[2];
  const float* W_in = (const float*)d_in[3];
  const float* b_in = (const float*)d_in[4];
  const float* qkv_w = (const float*)d_in[5];
  const float* r_net_w = (const float*)d_in[6];
  const float* o_w = (const float*)d_in[7];
  const float* ln1_g = (const float*)d_in[8];
  const float* ln1_b = (const float*)d_in[9];
  const float* ff_w1 = (const float*)d_in[10];
  const float* ff_b1 = (const float*)d_in[11];
  const float* ff_w2 = (const float*)d_in[12];
  const float* ff_b2 = (const float*)d_in[13];
  const float* ln2_g = (const float*)d_in[14];
  const float* ln2_b = (const float*)d_in[15];
  const float* W_dec = (const float*)d_in[16];
  const float* b_dec = (const float*)d_in[17];
  const float* r_w_bias = (const float*)d_in[18];
  const float* r_r_bias = (const float*)d_in[19];

  float* outp = (float*)d_out;
  float* pred = outp;
  float* mems_out = outp + (size_t)BATCH * NLEADS * SEGL;

  float* ws = (float*)d_ws;                                   // floats
  float* r_buf = ws;                                          // KLEN*256
  float* core = r_buf + (size_t)KLENL * DMODEL;               // 2048*256
  float* cat = core + (size_t)SEGL * BATCH * DMODEL;          // 4096*256
  float* headsb = cat + (size_t)KLENL * BATCH * DMODEL;       // 4096*768
  float* rk = headsb + (size_t)KLENL * BATCH * 768;           // 2048*256
  float* vecb = rk + (size_t)KLENL * 256;                     // 2048*256
  float* abuf = vecb + (size_t)SEGL * BATCH * DMODEL;         // 2048*256
  float* hidden = abuf + (size_t)SEGL * BATCH * DMODEL;       // 2048*1024

  k_pos_emb<<<KLENL, DMODEL, 0, stream>>>(r_buf);
  k_input_proj<<<SEGL * BATCH, DMODEL, 0, stream>>>(src, W_in, b_in, core,
                                                    mems_out);
  for (int i = 0; i < LNUM; ++i) {
    const int ncat1 = MEML * BATCH * DMODEL;
    k_copy<<<(ncat1 + 255) / 256, 256, 0, stream>>>(
        cat, mems + (size_t)i * ncat1, ncat1);
    k_copy<<<(ncat1 + 255) / 256, 256, 0, stream>>>(cat + ncat1, core,
                                                    SEGL * BATCH * DMODEL);
    k_gemm<<<dim3(768 / 64, (KLENL * BATCH) / 64), 256, 0, stream>>>(
        cat, qkv_w + (size_t)i * DMODEL * 768, nullptr, headsb,
        KLENL * BATCH, 768, DMODEL, 0);
    k_gemm<<<dim3(256 / 64, KLENL / 64), 256, 0, stream>>>(
        r_buf, r_net_w + (size_t)i * DMODEL * 256, nullptr, rk,
        KLENL, 256, DMODEL, 0);
    k_attn<<<BATCH * HEADS * (SEGL / 16), 32, 0, stream>>>(
        headsb, rk, r_w_bias, r_r_bias, attn_mask, vecb);
    k_gemm<<<dim3(DMODEL / 64, (SEGL * BATCH) / 64), 256, 0, stream>>>(
        vecb, o_w + (size_t)i * 256 * DMODEL, nullptr, abuf,
        SEGL * BATCH, DMODEL, 256, 0);
    k_add_ln<<<SEGL * BATCH, DMODEL, 0, stream>>>(
        core, abuf, ln1_g + i * DMODEL, ln1_b + i * DMODEL, core, nullptr);
    k_gemm<<<dim3(DINNER / 64, (SEGL * BATCH) / 64), 256, 0, stream>>>(
        core, ff_w1 + (size_t)i * DMODEL * DINNER, ff_b1 + (size_t)i * DINNER,
        hidden, SEGL * BATCH, DINNER, DMODEL, 1);
    k_gemm<<<dim3(DMODEL / 64, (SEGL * BATCH) / 64), 256, 0, stream>>>(
        hidden, ff_w2 + (size_t)i * DINNER * DMODEL, ff_b2 + (size_t)i * DMODEL,
        abuf, SEGL * BATCH, DMODEL, DINNER, 0);
    k_add_ln<<<SEGL * BATCH, DMODEL, 0, stream>>>(
        core, abuf, ln2_g + i * DMODEL, ln2_b + i * DMODEL, core,
        mems_out + (size_t)(i + 1) * MEML * BATCH * DMODEL);
  }
  k_decoder<<<(BATCH * NLEADS * SEGL + 255) / 256, 256, 0, stream>>>(
      core, W_dec, b_dec, pred);
}